// DynamicConv2D_31928786878960
// MI455X (gfx1250) — compile-verified
//
#include <hip/hip_runtime.h>

typedef __attribute__((ext_vector_type(2))) float v2f;
typedef __attribute__((ext_vector_type(8))) float v8f;

#define BATCH 32
#define CIN   64
#define CINP  66      // cin dim padded: 66 dwords -> LDS bank stride 2, conflict-free b64
#define COUT  64
#define IMG_H 256
#define IMG_W 256
#define TILE_X 128
#define COLS  130     // x-1 .. x+128
#define XT_PER_ROW (IMG_W / TILE_X)   // 2

// One block: group b, output row y, 128-pixel x tile, all 64 c_out.
// 8 waves: wave w -> c_out tile m = w%4, N-tiles n = (w/4) + 2*j, j=0..3.
__launch_bounds__(256, 1)
__global__ void conv_grouped_wmma_f32(const float* __restrict__ x,
                                      const float* __restrict__ kf,
                                      float* __restrict__ out)
{
    // layout: xs[(r*COLS + c)*CINP + cin] -> B fragment (cin, cin+1) is one b64
    __shared__ float xs[3 * COLS * CINP];   // 3*130*66*4 = 102,960 B

    const int block = blockIdx.x;
    const int xt = block % XT_PER_ROW;
    const int y  = (block / XT_PER_ROW) % IMG_H;
    const int b  = block / (XT_PER_ROW * IMG_H);
    const int xbase = xt * TILE_X;

    const int tid = threadIdx.x;

    // ---- Stage input halo tile into LDS: rows y-1..y+1, cols xbase-1..xbase+128
    const float* xg = x + (size_t)b * CIN * IMG_H * IMG_W;
    for (int i = tid; i < CIN * 3 * COLS; i += 256) {
        int c  = i % COLS;                  // consecutive tid -> consecutive gx (coalesced)
        int r  = (i / COLS) % 3;
        int ci = i / (COLS * 3);
        int gy = y + r - 1;
        int gx = xbase + c - 1;
        float v = 0.0f;
        if ((unsigned)gy < IMG_H && (unsigned)gx < IMG_W)
            v = xg[(size_t)ci * (IMG_H * IMG_W) + (size_t)gy * IMG_W + gx];
        xs[(r * COLS + c) * CINP + ci] = v;
    }
    __syncthreads();

    const int wave = tid >> 5;
    const int lane = tid & 31;
    const int lm   = lane & 15;     // M (A) / N (B,D) position
    const int hh   = lane >> 4;     // half-wave selector
    const int mtile = wave & 3;     // c_out tile (16 rows)
    const int npar  = wave >> 2;    // 0 or 1 -> N-tiles {0,2,4,6} or {1,3,5,7}

    v8f acc[4] = {};                // four 16x16 output tiles

    // Per-lane kernel row base: cout = mtile*16 + lm
    const float* kg = kf + ((size_t)b * COUT + mtile * 16 + lm) * (CIN * 9);

    for (int kh = 0; kh < 3; ++kh) {
        for (int kw = 0; kw < 3; ++kw) {
            const int tap = kh * 3 + kw;
            for (int cs = 0; cs < 16; ++cs) {
                const int cin0 = cs * 4 + 2 * hh;   // this lane's K pair base
                // A fragment (16x4 f32): VGPR0 = K0/K2, VGPR1 = K1/K3
                v2f a;
                a.x = kg[(size_t)cin0 * 9 + tap];
                a.y = kg[(size_t)(cin0 + 1) * 9 + tap];
#pragma unroll
                for (int j = 0; j < 4; ++j) {
                    const int nt  = npar + 2 * j;
                    const int col = nt * 16 + lm + kw;          // 0..129
                    // B fragment (4x16 f32): rows (cin0, cin0+1) -> single ds_load_b64
                    const v2f bfrag =
                        *(const v2f*)&xs[(kh * COLS + col) * CINP + cin0];
                    acc[j] = __builtin_amdgcn_wmma_f32_16x16x4_f32(
                        /*neg_a=*/false, a, /*neg_b=*/false, bfrag,
                        /*c_mod=*/(short)0, acc[j],
                        /*reuse_a=*/false, /*reuse_b=*/false);
                }
            }
        }
    }

    // ---- Store D tiles: VGPR r -> cout = r + 8*hh, N = lm
    float* og = out + ((size_t)b * COUT + mtile * 16) * (IMG_H * IMG_W)
                    + (size_t)y * IMG_W + xbase;
#pragma unroll
    for (int j = 0; j < 4; ++j) {
        const int nt = npar + 2 * j;
#pragma unroll
        for (int r = 0; r < 8; ++r) {
            og[(size_t)(r + 8 * hh) * (IMG_H * IMG_W) + nt * 16 + lm] = acc[j][r];
        }
    }
}

extern "C" void kernel_launch(void* const* d_in, const int* in_sizes, int n_in,
                              void* d_out, int out_size, void* d_ws, size_t ws_size,
                              hipStream_t stream) {
    const float* x  = (const float*)d_in[0];   // [1, B*CIN, H, W]
    const float* kf = (const float*)d_in[1];   // [B*COUT, CIN, 3, 3]
    float* out = (float*)d_out;                // [1, B*COUT, H, W]

    dim3 grid(BATCH * IMG_H * XT_PER_ROW);     // 32*256*2 = 16384 blocks
    dim3 blk(256);
    conv_grouped_wmma_f32<<<grid, blk, 0, stream>>>(x, kf, out);
}